// TopKPooling_33157147525935
// MI455X (gfx1250) — compile-verified
//
#include <hip/hip_runtime.h>
#include <hip/hip_bf16.h>

// ---------------- problem constants (from the reference) ----------------
#define NG    64          // graphs
#define NPG   4096        // nodes per graph
#define CH    256         // in channels
#define KSEL  2048        // ceil(0.5 * 4096)
#define NTOT  (NG * NPG)          // 262144 nodes
#define ETOT  (NG * NPG * 16)     // 4194304 edges
#define NKTOT (NG * KSEL)         // 131072 kept nodes

#define KCH        64     // K-chunk staged per async round
#define LDS_STRIDE 68     // 64 + 4 floats pad -> bank = (4*row + k) % 64, conflict-free

typedef __attribute__((ext_vector_type(2))) float v2f;
typedef __attribute__((ext_vector_type(8))) float v8f;

// ---------------- kernel 0: wn[k] = w[k] / ||w|| ----------------
__global__ void normalize_weight(const float* __restrict__ w,
                                 float* __restrict__ wn) {
    __shared__ float red[CH];
    int t = threadIdx.x;
    float v = w[t];
    red[t] = v * v;
    __syncthreads();
    for (int s = CH / 2; s > 0; s >>= 1) {
        if (t < s) red[t] += red[t + s];
        __syncthreads();
    }
    wn[t] = v * rsqrtf(red[0]);
}

// ---------------- kernel 1: score = tanh(x . wn) --------------------------
// One wave -> one 16-row tile. Per K-chunk of 64:
//   1) stage the 16x64 f32 chunk into LDS with GLOBAL_LOAD_ASYNC_TO_LDS_B128
//      (coalesced: each lane moves 16 B; one instruction covers two
//      contiguous 256 B row segments), ASYNCcnt-tracked;
//   2) s_wait_asynccnt 0, then 16 accumulating V_WMMA_F32_16X16X4_F32 with
//      A fragments from LDS (conflict-free banks) and B = wn in column N=0.
// Column 0 of the C/D tile then holds the 16 dot products (ISA 7.12.2):
// lane 0 VGPRs 0..7 = rows 0..7, lane 16 VGPRs 0..7 = rows 8..15.
__global__ void __launch_bounds__(256) score_wmma(
        const float* __restrict__ x, const float* __restrict__ wn,
        float* __restrict__ score) {
    __shared__ float tile[8 * 16 * LDS_STRIDE];   // 34,816 B, 8 waves
    int wave = threadIdx.x >> 5;
    int lane = threadIdx.x & 31;
    int half = lane >> 4;    // 0: K=0,1   1: K=2,3 (A/B lane striping)
    int m    = lane & 15;    // A row / B column
    int tileBase = blockIdx.x * 128 + wave * 16;

    float* wtile = &tile[wave * 16 * LDS_STRIDE];
    // staging coordinates: instruction i covers rows (half + 2*i),
    // columns (m*4 .. m*4+3) of the 16x64 chunk
    int scol = m * 4;

    v8f c = {};
    for (int kc = 0; kc < CH; kc += KCH) {
        // ensure pending ds_loads from the previous chunk have landed in
        // VGPRs before the DMA overwrites the LDS tile
        asm volatile("s_wait_dscnt 0x0" ::: "memory");
#pragma unroll
        for (int i = 0; i < 8; ++i) {
            int row = half + 2 * i;
            unsigned long long gsrc = (unsigned long long)(uintptr_t)(
                x + (size_t)(tileBase + row) * CH + kc + scol);
            unsigned ldst = (unsigned)(uintptr_t)&wtile[row * LDS_STRIDE + scol];
            asm volatile("global_load_async_to_lds_b128 %0, %1, off"
                         :: "v"(ldst), "v"(gsrc) : "memory");
        }
        asm volatile("s_wait_asynccnt 0x0" ::: "memory");

        const float* xr = &wtile[m * LDS_STRIDE];
#pragma unroll
        for (int k0 = 0; k0 < KCH; k0 += 4) {
            int kk = k0 + 2 * half;
            v2f a, b;
            a[0] = xr[kk];
            a[1] = xr[kk + 1];
            float b0 = wn[kc + kk];
            float b1 = wn[kc + kk + 1];
            b[0] = (m == 0) ? b0 : 0.0f;   // only column N=0 carries the weight
            b[1] = (m == 0) ? b1 : 0.0f;
            c = __builtin_amdgcn_wmma_f32_16x16x4_f32(
                    /*neg_a=*/false, a, /*neg_b=*/false, b,
                    /*c_mod=*/(short)0, c, /*reuse_a=*/false, /*reuse_b=*/false);
        }
    }
    if (m == 0) {
        int rbase = tileBase + half * 8;
#pragma unroll
        for (int j = 0; j < 8; ++j)
            score[rbase + j] = tanhf(c[j]);
    }
}

// ---------------- kernel 2: new_id = -1 ----------------
__global__ void fill_neg1(int* __restrict__ p) {
    p[blockIdx.x * 256 + threadIdx.x] = -1;
}

// ---------------- kernel 3: per-graph descending bitonic sort + select -----
// One workgroup per graph; 4096 (score,idx) pairs in 32 KB LDS.
// Order matches jax.lax.top_k: descending score, ties -> lower index first.
__global__ void __launch_bounds__(512) topk_sort(
        const float* __restrict__ score, int* __restrict__ new_id,
        int* __restrict__ perm, float* __restrict__ score_perm,
        int* __restrict__ batch_out) {
    __shared__ float sK[NPG];
    __shared__ int   sI[NPG];
    int g = blockIdx.x;
    int t = threadIdx.x;
    for (int i = t; i < NPG; i += 512) {
        sK[i] = score[g * NPG + i];
        sI[i] = i;
    }
    __syncthreads();
    for (int k = 2; k <= NPG; k <<= 1) {
        for (int j = k >> 1; j > 0; j >>= 1) {
            for (int i = t; i < NPG; i += 512) {
                int ixj = i ^ j;
                if (ixj > i) {
                    float s1 = sK[i], s2 = sK[ixj];
                    int   a1 = sI[i], a2 = sI[ixj];
                    bool pre = (s1 > s2) || (s1 == s2 && a1 < a2); // i precedes ixj
                    bool up  = ((i & k) == 0);
                    if (up ? !pre : pre) {
                        sK[i] = s2; sK[ixj] = s1;
                        sI[i] = a2; sI[ixj] = a1;
                    }
                }
            }
            __syncthreads();
        }
    }
    for (int j = t; j < KSEL; j += 512) {
        int gid = g * NPG + sI[j];   // global node id
        int out = g * KSEL + j;      // new node id
        perm[out]       = gid;
        score_perm[out] = sK[j];
        batch_out[out]  = g;
        new_id[gid]     = out;
    }
}

// ---------------- kernel 4: x_out[r] = x[perm[r]] * MULTIPLIER -------------
__global__ void __launch_bounds__(256) gather_rows(
        const float* __restrict__ x, const int* __restrict__ perm,
        float* __restrict__ xout) {
    int r = blockIdx.x * 4 + (threadIdx.x >> 6);  // 4 rows / block
    int t = threadIdx.x & 63;                     // 64 lanes x float4 = 256
    const float4* src = (const float4*)(x + (size_t)perm[r] * CH);
    float4* dst = (float4*)(xout + (size_t)r * CH);
    float4 v = src[t];
    const float mult = 1.0f;  // MULTIPLIER
    v.x *= mult; v.y *= mult; v.z *= mult; v.w *= mult;
    dst[t] = v;
}

// ---------------- kernel 5: relabel + filter edges -------------------------
__global__ void __launch_bounds__(256) filter_edges(
        const int* __restrict__ ei, const float* __restrict__ ea,
        const int* __restrict__ new_id,
        int* __restrict__ ei_out, float* __restrict__ ea_out,
        unsigned char* __restrict__ mask_out) {
    long long e = (long long)blockIdx.x * 256 + threadIdx.x;
    int r = ei[e];
    int c = ei[(long long)ETOT + e];
    int nr = new_id[r];
    int nc = new_id[c];
    bool keep = (nr >= 0) && (nc >= 0);
    ei_out[e]                    = keep ? nr : -1;
    ei_out[(long long)ETOT + e]  = keep ? nc : -1;
    ea_out[e]                    = keep ? ea[e] : 0.0f;
    mask_out[e]                  = keep ? (unsigned char)1 : (unsigned char)0;
}

// ---------------- launcher -------------------------------------------------
extern "C" void kernel_launch(void* const* d_in, const int* in_sizes, int n_in,
                              void* d_out, int out_size, void* d_ws, size_t ws_size,
                              hipStream_t stream) {
    // inputs per setup_inputs() order: x, edge_index, edge_attr, batch, weight
    const float* x  = (const float*)d_in[0];
    const int*   ei = (const int*)  d_in[1];
    const float* ea = (const float*)d_in[2];
    // d_in[3] (batch) not needed: batch_out is derivable from graph id
    const float* w  = (const float*)d_in[4];

    // workspace layout (~2.1 MB): wn[256] | score[NTOT] | new_id[NTOT]
    float* wn     = (float*)d_ws;
    float* score  = wn + CH;
    int*   new_id = (int*)(score + NTOT);

    // d_out: reference tuple concatenated flat in return order
    //   x_out[NKTOT*CH] f32 | edge_index_out[2*E] i32 | edge_attr_out[E] f32 |
    //   batch_out[NKTOT] i32 | perm[NKTOT] i32 | score_perm[NKTOT] f32 |
    //   edge_mask[E] bool(1B)
    char*  base       = (char*)d_out;
    float* x_out      = (float*)base;
    int*   ei_out     = (int*)  (base + (size_t)NKTOT * CH * 4);
    float* ea_out     = (float*)((char*)ei_out  + (size_t)2 * ETOT * 4);
    int*   batch_out  = (int*)  ((char*)ea_out  + (size_t)ETOT * 4);
    int*   perm       = (int*)  ((char*)batch_out + (size_t)NKTOT * 4);
    float* score_perm = (float*)((char*)perm      + (size_t)NKTOT * 4);
    unsigned char* emask = (unsigned char*)((char*)score_perm + (size_t)NKTOT * 4);

    normalize_weight<<<1, CH, 0, stream>>>(w, wn);
    score_wmma<<<NTOT / 128, 256, 0, stream>>>(x, wn, score);
    fill_neg1<<<NTOT / 256, 256, 0, stream>>>(new_id);
    topk_sort<<<NG, 512, 0, stream>>>(score, new_id, perm, score_perm, batch_out);
    gather_rows<<<NKTOT / 4, 256, 0, stream>>>(x, perm, x_out);
    filter_edges<<<ETOT / 256, 256, 0, stream>>>(ei, ea, new_id, ei_out, ea_out, emask);
}